// Edge2Node_6691559047520
// MI455X (gfx1250) — compile-verified
//
#include <hip/hip_runtime.h>
#include <hip/hip_bf16.h>

typedef __attribute__((ext_vector_type(2))) float v2f;
typedef __attribute__((ext_vector_type(8))) float v8f;

#define WT_STRIDE 66  // even pad: 8B-aligned float2 column reads, spreads banks

// ---------------------------------------------------------------------------
// Kernel 1: zero workspace
// ---------------------------------------------------------------------------
__global__ void zero_kernel(float* __restrict__ p, long long n) {
    long long i = (long long)blockIdx.x * blockDim.x + threadIdx.x;
    long long stride = (long long)gridDim.x * blockDim.x;
    for (; i < n; i += stride) p[i] = 0.0f;
}

// ---------------------------------------------------------------------------
// Kernel 2: per-node degree counts (float, matching reference semantics)
// ---------------------------------------------------------------------------
__global__ void count_kernel(const int* __restrict__ eidx,
                             float* __restrict__ cntS,
                             float* __restrict__ cntD, int E) {
    int e = blockIdx.x * blockDim.x + threadIdx.x;
    if (e < E) {
        unsafeAtomicAdd(&cntS[eidx[e]], 1.0f);
        unsafeAtomicAdd(&cntD[eidx[E + e]], 1.0f);
    }
}

// ---------------------------------------------------------------------------
// Per-tile body: WMMA fp32 GEMM (16 edges x 64 cols) + ReLU + atomic scatter.
// FULL=true  -> all 16 rows valid: no divergent guards anywhere (EXEC stays
//               all-1s; atomics issue as a straight clause).
// FULL=false -> tail tile: per-row guards.
//   A: 16x4 f32  -> v2f per lane: lanes 0-15 K=kb,kb+1 ; lanes 16-31 K=kb+2,kb+3
//   C: 16x16 f32 -> v8f per lane: VGPR r, lanes 0-15 M=r ; lanes 16-31 M=r+8
// ---------------------------------------------------------------------------
template <bool FULL>
__device__ __forceinline__ void process_tile(
    const float* __restrict__ emb, const int* __restrict__ eidx,
    float* __restrict__ accS, float* __restrict__ accD,
    const float* sWt, const float* sBias,
    int E, int rowbase, int half, int l16, int kbase) {

    // ---- load 16 A fragments as float2 (covers K = 0..63 in steps of 4) ----
    v2f a[16];
    const int  arow = rowbase + l16;
    const v2f* ap   = (const v2f*)(emb + (long long)arow * 64 + kbase);
    if (FULL) {
#pragma unroll
        for (int j = 0; j < 16; ++j) a[j] = ap[2 * j];
    } else {
        const bool rowok = (arow < E);
#pragma unroll
        for (int j = 0; j < 16; ++j) {
            if (rowok) a[j] = ap[2 * j];
            else       a[j] = (v2f){0.0f, 0.0f};
        }
    }

    // ---- edge endpoints for the 8 C-rows this lane holds ----
    int srcs[8], dsts[8];
#pragma unroll
    for (int r = 0; r < 8; ++r) {
        int m  = rowbase + r + 8 * half;
        int mm = FULL ? m : ((m < E) ? m : 0);
        srcs[r] = eidx[mm];
        dsts[r] = eidx[E + mm];
    }

    // ---- 4 output column blocks of 16; keep ROLLED to bound VGPR pressure ----
#pragma unroll 1
    for (int nb = 0; nb < 4; ++nb) {
        const int col = nb * 16 + l16;

        // preload this block's 16 B fragments from LDS (ds_load_2addr_b64 pairs)
        v2f bfr[16];
        const v2f* wp = (const v2f*)(sWt + col * WT_STRIDE + kbase);
#pragma unroll
        for (int j = 0; j < 16; ++j) {
            bfr[j] = wp[2 * j];  // B[K=4j+kbase][col], B[K=4j+kbase+1][col]
        }

        const float bval = sBias[col];
        v8f acc = {bval, bval, bval, bval, bval, bval, bval, bval};

#pragma unroll
        for (int j = 0; j < 16; ++j) {
            acc = __builtin_amdgcn_wmma_f32_16x16x4_f32(
                /*neg_a=*/false, a[j], /*neg_b=*/false, bfr[j],
                /*c_mod=*/(short)0, acc, /*reuse_a=*/false, /*reuse_b=*/false);
        }

        // ---- ReLU + coalesced atomic scatter into both accumulators ----
#pragma unroll
        for (int r = 0; r < 8; ++r) {
            if (FULL) {
                float v = fmaxf(acc[r], 0.0f);
                unsafeAtomicAdd(&accS[(long long)srcs[r] * 64 + col], v);
                unsafeAtomicAdd(&accD[(long long)dsts[r] * 64 + col], v);
            } else {
                int m = rowbase + r + 8 * half;
                if (m < E) {
                    float v = fmaxf(acc[r], 0.0f);
                    unsafeAtomicAdd(&accS[(long long)srcs[r] * 64 + col], v);
                    unsafeAtomicAdd(&accD[(long long)dsts[r] * 64 + col], v);
                }
            }
        }
    }
}

// ---------------------------------------------------------------------------
// Kernel 3 driver
// ---------------------------------------------------------------------------
__global__ void edge_proj_scatter(const float* __restrict__ emb,
                                  const int* __restrict__ eidx,
                                  const float* __restrict__ W,
                                  const float* __restrict__ bias,
                                  float* __restrict__ accS,
                                  float* __restrict__ accD,
                                  int E, int ntiles) {
    __shared__ float sWt[64 * WT_STRIDE];  // W transposed: sWt[n*66 + k] = W[k*64+n]
    __shared__ float sBias[64];

    const int tid = threadIdx.x;
    for (int i = tid; i < 64 * 64; i += blockDim.x) {
        int k = i >> 6;
        int n = i & 63;
        sWt[n * WT_STRIDE + k] = W[i];
    }
    if (tid < 64) sBias[tid] = bias[tid];
    __syncthreads();

    const int lane = tid & 31;
    const int wave = tid >> 5;
    const int wavesPerBlock = blockDim.x >> 5;
    const int gwave = blockIdx.x * wavesPerBlock + wave;
    const int nwaves = gridDim.x * wavesPerBlock;

    const int half  = lane >> 4;   // 0 or 1
    const int l16   = lane & 15;
    const int kbase = half * 2;    // fp32 WMMA A/B layout: K pair offset per half-wave

    for (int tile = gwave; tile < ntiles; tile += nwaves) {
        const int rowbase = tile * 16;
        if (rowbase + 16 <= E) {   // wave-uniform branch: EXEC untouched
            process_tile<true>(emb, eidx, accS, accD, sWt, sBias,
                               E, rowbase, half, l16, kbase);
        } else {
            process_tile<false>(emb, eidx, accS, accD, sWt, sBias,
                                E, rowbase, half, l16, kbase);
        }
    }
}

// ---------------------------------------------------------------------------
// Kernel 4: out = 0.5 * (accS/max(cS,1) + accD/max(cD,1))
// ---------------------------------------------------------------------------
__global__ void finalize_kernel(const float* __restrict__ accS,
                                const float* __restrict__ accD,
                                const float* __restrict__ cntS,
                                const float* __restrict__ cntD,
                                float* __restrict__ out, long long total) {
    long long i = (long long)blockIdx.x * blockDim.x + threadIdx.x;
    if (i < total) {
        int n = (int)(i >> 6);
        float s = accS[i] / fmaxf(cntS[n], 1.0f);
        float d = accD[i] / fmaxf(cntD[n], 1.0f);
        out[i] = 0.5f * (s + d);
    }
}

// ---------------------------------------------------------------------------
extern "C" void kernel_launch(void* const* d_in, const int* in_sizes, int n_in,
                              void* d_out, int out_size, void* d_ws, size_t ws_size,
                              hipStream_t stream) {
    const float* emb  = (const float*)d_in[0];   // [E,64]
    const int*   eidx = (const int*)d_in[1];     // [2,E]
    const float* W    = (const float*)d_in[3];   // [64,64]
    const float* bias = (const float*)d_in[4];   // [64]
    float*       out  = (float*)d_out;           // [N,64]

    const int E = in_sizes[0] / 64;
    const int N = out_size / 64;

    // workspace layout (floats): accS[N*64] | accD[N*64] | cntS[N] | cntD[N]
    float* accS = (float*)d_ws;
    float* accD = accS + (long long)N * 64;
    float* cntS = accD + (long long)N * 64;
    float* cntD = cntS + N;

    const long long zn = (long long)N * 128 + 2LL * N;
    {
        int blocks = (int)((zn + 255) / 256);
        if (blocks > 4096) blocks = 4096;
        zero_kernel<<<blocks, 256, 0, stream>>>((float*)d_ws, zn);
    }

    count_kernel<<<(E + 255) / 256, 256, 0, stream>>>(eidx, cntS, cntD, E);

    const int ntiles = (E + 15) / 16;
    {
        const int wavesPerBlock = 4;       // 128 threads = 4 wave32
        int blocks = (ntiles + wavesPerBlock - 1) / wavesPerBlock;
        if (blocks > 16384) blocks = 16384;  // grid-stride covers the rest
        edge_proj_scatter<<<blocks, 32 * wavesPerBlock, 0, stream>>>(
            emb, eidx, W, bias, accS, accD, E, ntiles);
    }

    {
        long long total = (long long)N * 64;
        finalize_kernel<<<(int)((total + 255) / 256), 256, 0, stream>>>(
            accS, accD, cntS, cntD, out, total);
    }
}